// ProbSparseAttention_35347580846623
// MI455X (gfx1250) — compile-verified
//
#include <hip/hip_runtime.h>
#include <hip/hip_bf16.h>

typedef __attribute__((ext_vector_type(2))) float v2f;
typedef __attribute__((ext_vector_type(8))) float v8f;

#define HEADS   8
#define SEQ     4096
#define DIM     64
#define TOPK    81          // max(1, int(4096 * 0.02))
#define QTILE   16
#define SROW    4100        // padded LDS row stride (floats): +4 breaks bank conflicts
#define SEL_PAD 96
#define NWAVES  8
#define KEYS_PER_WAVE (SEQ / NWAVES)        // 512
#define KT_PER_WAVE   (KEYS_PER_WAVE / 16)  // 32 key tiles per wave

// Order-preserving float->uint map (branch-free: 3 VALU)
__device__ __forceinline__ unsigned f2ord(float f) {
    unsigned b = __float_as_uint(f);
    return b ^ ((unsigned)(((int)b) >> 31) | 0x80000000u);
}
__device__ __forceinline__ float ord2f(unsigned u) {
    unsigned b = u ^ ((unsigned)(((int)~u) >> 31) | 0x80000000u);
    return __uint_as_float(b);
}
__device__ __forceinline__ unsigned wred_max_u(unsigned v) {
#pragma unroll
    for (int m = 16; m >= 1; m >>= 1) {
        unsigned o = (unsigned)__shfl_xor((int)v, m, 32);
        v = (o > v) ? o : v;
    }
    return v;
}
__device__ __forceinline__ float wred_add_f(float v) {
#pragma unroll
    for (int m = 16; m >= 1; m >>= 1) v += __shfl_xor(v, m, 32);
    return v;
}

__global__ __launch_bounds__(256, 1)
void probsparse_attn_kernel(const float* __restrict__ Q,
                            const float* __restrict__ K,
                            const float* __restrict__ V,
                            float* __restrict__ O) {
    extern __shared__ char smem[];
    float* S      = (float*)smem;                                          // [16][4100]
    int*   selIdx = (int*)  (smem + (size_t)QTILE * SROW * sizeof(float)); // [16][96]
    float* selP   = (float*)((char*)selIdx + (size_t)QTILE * SEL_PAD * sizeof(int));

    const int lane = threadIdx.x & 31;
    const int wave = threadIdx.x >> 5;
    const int head = blockIdx.x / (SEQ / QTILE);
    const int q0   = (blockIdx.x % (SEQ / QTILE)) * QTILE;

    const float* Qh = Q + (size_t)head * SEQ * DIM;
    const float* Kh = K + (size_t)head * SEQ * DIM;
    const float* Vh = V + (size_t)head * SEQ * DIM;
    float*       Oh = O + (size_t)head * SEQ * DIM;

    // ---------------- Phase 1: S[16][4096] = (Q_tile . K^T) * 0.125 via f32 WMMA -------------
    // A (16x4 f32): lanes 0-15 -> M=lane, {K0,K1}; lanes 16-31 -> {K2,K3}. B mirrors with N on lanes.
    // C/D: 8 VGPRs, N = lane%16, M = r + 8*(lane/16).
    const int half = lane >> 4;
    const int mrow = lane & 15;

    v2f a[16];
#pragma unroll
    for (int kk = 0; kk < 16; ++kk) {
        a[kk] = *(const v2f*)(Qh + (size_t)(q0 + mrow) * DIM + kk * 4 + 2 * half);
    }

    for (int t = 0; t < KT_PER_WAVE; ++t) {
        const int n0 = wave * KEYS_PER_WAVE + t * 16;
        v8f c = {0.f, 0.f, 0.f, 0.f, 0.f, 0.f, 0.f, 0.f};
#pragma unroll
        for (int kk = 0; kk < 16; ++kk) {
            // B[kd][n] = K[n0+n][kk*4+kd]   (scores = Q . K^T)
            v2f b = *(const v2f*)(Kh + (size_t)(n0 + mrow) * DIM + kk * 4 + 2 * half);
            c = __builtin_amdgcn_wmma_f32_16x16x4_f32(
                    false, a[kk], false, b, (short)0, c, false, false);
        }
#pragma unroll
        for (int r = 0; r < 8; ++r) {
            const int row = r + 8 * half;
            S[(size_t)row * SROW + n0 + mrow] = c[r] * 0.125f;  // 1/sqrt(64)
        }
    }
    __syncthreads();

    // ---------------- Phases 2+3: each wave owns rows (2*wave, 2*wave+1) ----------------------
#pragma unroll
    for (int rr = 0; rr < 2; ++rr) {
        const int row = wave * 2 + rr;
        const float* Srow = &S[(size_t)row * SROW];

        // 128 values/lane as order-preserving uints (ds_load_b128)
        unsigned u[128];
#pragma unroll
        for (int i = 0; i < 32; ++i) {
            float4 f4 = *(const float4*)(Srow + lane * 4 + i * 128);
            u[4 * i + 0] = f2ord(f4.x);
            u[4 * i + 1] = f2ord(f4.y);
            u[4 * i + 2] = f2ord(f4.z);
            u[4 * i + 3] = f2ord(f4.w);
        }

        // Row max (softmax shift)
        unsigned um = 0;
#pragma unroll
        for (int i = 0; i < 128; ++i) um = (u[i] > um) ? u[i] : um;
        um = wred_max_u(um);
        const float m = ord2f(um);

        // Exact 81st-largest: scalar binary search; counts via ballot -> SGPR popcount chain
        // (v_cmp on VALU, popcount/add on SALU co-issue; lo/hi/branches fully scalar)
        unsigned lo = 0u, hi = 0xFFFFFFFFu;
        while (lo < hi) {
            unsigned mid = (unsigned)(((unsigned long long)lo + (unsigned long long)hi + 1ull) >> 1);
            int c = 0;
#pragma unroll
            for (int i = 0; i < 128; ++i) {
                c += __builtin_popcount(__builtin_amdgcn_ballot_w32(u[i] >= mid));
            }
            if (c >= TOPK) lo = mid; else hi = mid - 1;
        }
        const unsigned thr = lo;   // count(>= thr) == TOPK (scores bit-distinct)

        // Atomic-free compaction: ballot prefix ranks, SGPR running base.
        // Deterministic list order (by register slot i, then by lane).
        const unsigned lane_lt = (1u << lane) - 1u;
        int base = 0;
        float lsum = 0.f;
#pragma unroll
        for (int i = 0; i < 128; ++i) {
            const bool hit = (u[i] >= thr);
            const unsigned bal = __builtin_amdgcn_ballot_w32(hit);
            if (hit) {
                const int j = (i >> 2) * 128 + lane * 4 + (i & 3);  // key index in row
                const float p = __expf(ord2f(u[i]) - m);
                lsum += p;
                const int pos = base + __builtin_popcount(bal & lane_lt);
                selIdx[row * SEL_PAD + pos] = j;
                selP[row * SEL_PAD + pos]   = p;
            }
            base += __builtin_popcount(bal);
        }
        const int n = base;                     // == TOPK
        const float inv = 1.0f / wred_add_f(lsum);

        // Sparse P.V: lane owns output dims {lane, lane+32}; only 81 V rows touched
        float a0 = 0.f, a1 = 0.f;
        for (int jj = 0; jj < n; ++jj) {
            const int   idx = selIdx[row * SEL_PAD + jj];
            const float p   = selP[row * SEL_PAD + jj];
            const float* Vr = Vh + (size_t)idx * DIM;
            a0 += p * Vr[lane];
            a1 += p * Vr[lane + 32];
        }
        float* Orow = Oh + (size_t)(q0 + row) * DIM;
        Orow[lane]      = a0 * inv;
        Orow[lane + 32] = a1 * inv;
    }
}

extern "C" void kernel_launch(void* const* d_in, const int* in_sizes, int n_in,
                              void* d_out, int out_size, void* d_ws, size_t ws_size,
                              hipStream_t stream) {
    (void)in_sizes; (void)n_in; (void)out_size; (void)d_ws; (void)ws_size;
    const float* Q = (const float*)d_in[0];
    const float* K = (const float*)d_in[1];
    const float* V = (const float*)d_in[2];
    float* O = (float*)d_out;

    const size_t smem = (size_t)QTILE * SROW * sizeof(float)     // 262,400 B scores
                      + (size_t)QTILE * SEL_PAD * sizeof(int)    //   6,144 B indices
                      + (size_t)QTILE * SEL_PAD * sizeof(float); //   6,144 B probs  => 274,688 B < 320 KB

    hipFuncSetAttribute((const void*)probsparse_attn_kernel,
                        hipFuncAttributeMaxDynamicSharedMemorySize, (int)smem);

    dim3 grid(HEADS * (SEQ / QTILE));   // 2048 workgroups
    dim3 block(256);                    // 8 waves (wave32)
    probsparse_attn_kernel<<<grid, block, smem, stream>>>(Q, K, V, O);
}